// SubjectPlusLayer_83056077570346
// MI455X (gfx1250) — compile-verified
//
#include <hip/hip_runtime.h>

// CDNA5 / gfx1250 WMMA fragment types (V_WMMA_F32_16X16X4_F32: A/B = 2 VGPR, C/D = 8 VGPR)
typedef __attribute__((ext_vector_type(2))) float v2f;
typedef __attribute__((ext_vector_type(8))) float v8f;

#define N_SUBJ 100
#define DIM    270              // N_OUT == N_IN == 270
#define TLEN   3000
#define LDS_STRIDE 274          // even (8B-aligned b64) ; 274%64=18 -> bank-skewed rows
#define MT_PER_BLK 6            // m-tiles per block (3 groups cover 17 tiles + 1 phantom)
#define ROWS_PER_BLK (MT_PER_BLK * 16)   // 96
#define F2_PER_ROW 136          // 272 floats staged per row (cols 270/271 zero pad)

__global__ __launch_bounds__(256)
void subject_gemm_wmma(const float* __restrict__ x,
                       const long long* __restrict__ s,
                       const float* __restrict__ A,
                       float* __restrict__ out)
{
    const int nChunk = blockIdx.x;   // 24 chunks of 128 columns
    const int mGroup = blockIdx.y;   // 3 groups of 6 m-tiles (96 rows)
    const int b      = blockIdx.z;   // batch

    const int tid  = threadIdx.x;
    const int wave = tid >> 5;
    const int lane = tid & 31;

    long long sel = s[b];
    if (sel < 0 || sel > N_SUBJ) sel = 0;   // reference: slot 0 is the zero matrix

    __shared__ float ldsA[ROWS_PER_BLK * LDS_STRIDE];   // ~105 KB of the 320 KB pool

    const int r0 = mGroup * ROWS_PER_BLK;
    const float* Ag = (sel > 0) ? (A + (size_t)(sel - 1) * DIM * DIM) : A;

    // Stage A[sel][r0 .. r0+95][0..271] into LDS (zero padded), float2-vectorized.
    for (int idx = tid; idx < ROWS_PER_BLK * F2_PER_ROW; idx += 256) {
        const int r  = idx / F2_PER_ROW;
        const int c2 = idx - r * F2_PER_ROW;
        const int gr = r0 + r;
        v2f v = {0.0f, 0.0f};
        if (sel > 0 && gr < DIM && 2 * c2 < DIM)        // DIM even -> pairs never straddle
            v = *(const v2f*)&Ag[(size_t)gr * DIM + 2 * c2];
        *(v2f*)&ldsA[r * LDS_STRIDE + 2 * c2] = v;
    }
    __syncthreads();

    const int col     = lane & 15;   // N index within wave strip / A-matrix M row per lane
    const int halfSel = lane >> 4;   // 0: lanes 0-15 (K, K+1), 1: lanes 16-31 (K+2, K+3)
    const int n  = nChunk * 128 + wave * 16 + col;
    const int nc = (n < TLEN) ? n : (TLEN - 1);         // clamped load column (no exec dance)

    // Per-lane base into x[b]: row 2*halfSel, column nc.
    const float* XbL = x + (size_t)b * DIM * TLEN + (size_t)(2 * halfSel) * TLEN + nc;

    v8f acc[MT_PER_BLK];
    #pragma unroll
    for (int j = 0; j < MT_PER_BLK; ++j) acc[j] = (v8f){};

    const float* ldsRow = &ldsA[col * LDS_STRIDE];      // per-lane A row base

    // Main K loop: k = 0..267, completely unguarded.
    // One shared B fragment feeds 6 independent WMMA chains.
    #pragma unroll 2
    for (int k0 = 0; k0 < 268; k0 += 4) {
        v2f bf;
        bf.x = XbL[(size_t)k0 * TLEN];                  // X[k0 + 2*halfSel    ][nc]
        bf.y = XbL[(size_t)k0 * TLEN + TLEN];           // X[k0 + 2*halfSel + 1][nc]
        #pragma unroll
        for (int j = 0; j < MT_PER_BLK; ++j) {
            v2f a = *(const v2f*)&ldsRow[j * 16 * LDS_STRIDE + k0 + 2 * halfSel];
            acc[j] = __builtin_amdgcn_wmma_f32_16x16x4_f32(false, a, false, bf,
                                                           (short)0, acc[j], false, false);
        }
    }

    // Tail k0 = 268 (k = 268..271): clamp B row index; A cols 270/271 are zero in LDS.
    {
        const int kA  = 268 + 2 * halfSel;
        const int kb0 = (kA     < DIM) ? kA     : (DIM - 1);
        const int kb1 = (kA + 1 < DIM) ? kA + 1 : (DIM - 1);
        const float* Xb = x + (size_t)b * DIM * TLEN;
        v2f bf;
        bf.x = Xb[(size_t)kb0 * TLEN + nc];
        bf.y = Xb[(size_t)kb1 * TLEN + nc];
        #pragma unroll
        for (int j = 0; j < MT_PER_BLK; ++j) {
            v2f a = *(const v2f*)&ldsRow[j * 16 * LDS_STRIDE + 268 + 2 * halfSel];
            acc[j] = __builtin_amdgcn_wmma_f32_16x16x4_f32(false, a, false, bf,
                                                           (short)0, acc[j], false, false);
        }
    }

    // Store: C VGPR v -> row r0 + j*16 + v + 8*halfSel, column n (64B contiguous per half-wave).
    if (n < TLEN) {
        float* Ob = out + (size_t)b * DIM * TLEN + n;
        #pragma unroll
        for (int j = 0; j < MT_PER_BLK; ++j) {
            #pragma unroll
            for (int v = 0; v < 8; ++v) {
                const int r = r0 + j * 16 + v + 8 * halfSel;
                if (r < DIM)
                    Ob[(size_t)r * TLEN] = acc[j][v];
            }
        }
    }
}

extern "C" void kernel_launch(void* const* d_in, const int* in_sizes, int n_in,
                              void* d_out, int out_size, void* d_ws, size_t ws_size,
                              hipStream_t stream) {
    // setup_inputs() order: x (f32 [128,270,3000]), s (int64 [128]), A (f32 [100,270,270])
    const float*     x = (const float*)d_in[0];
    const long long* s = (const long long*)d_in[1];
    const float*     A = (const float*)d_in[2];
    float*         out = (float*)d_out;

    dim3 block(256, 1, 1);
    dim3 grid((TLEN + 127) / 128,   // 24 n-chunks
              3,                    // 3 m-groups of 6 tiles
              128);                 // batch
    subject_gemm_wmma<<<grid, block, 0, stream>>>(x, s, A, out);
}